// GCN_8761733284234
// MI455X (gfx1250) — compile-verified
//
#include <hip/hip_runtime.h>

typedef __attribute__((ext_vector_type(2))) float v2f;
typedef __attribute__((ext_vector_type(8))) float v8f;

#define NFEAT 128
#define NHID  128

__device__ __forceinline__ float sigmoidf_fast(float x) {
    return 1.0f / (1.0f + __expf(-x));
}

__device__ __forceinline__ void fatomic_add(float* p, float v) {
    // emits global_atomic_add_f32 (no CAS loop)
    unsafeAtomicAdd(p, v);
}

// ---------------------------------------------------------------- init ----
__global__ __launch_bounds__(256) void gcn_init_deg(float* __restrict__ deg, int N) {
    int i = blockIdx.x * blockDim.x + threadIdx.x;
    if (i < N) deg[i] = 1.0f;  // self-loop contributes weight 1.0
}

__global__ __launch_bounds__(256) void gcn_init_out_bias(float* __restrict__ out,
                                                         const float* __restrict__ b,
                                                         long long total) {
    long long i = (long long)blockIdx.x * blockDim.x + threadIdx.x;
    if (i < total) out[i] = b[(int)(i & (NHID - 1))];
}

// ------------------------------------------------------------- degrees ----
__global__ __launch_bounds__(256) void gcn_degree(const long long* __restrict__ col,
                                                  const float* __restrict__ ew,
                                                  float* __restrict__ deg, int E) {
    int e = blockIdx.x * blockDim.x + threadIdx.x;
    if (e >= E) return;
    float s = sigmoidf_fast(ew[e]);
    fatomic_add(&deg[(int)col[e]], s);
}

__global__ __launch_bounds__(256) void gcn_rsqrt(float* __restrict__ deg, int N) {
    int i = blockIdx.x * blockDim.x + threadIdx.x;
    if (i < N) {
        float d = deg[i];
        deg[i] = (d > 0.0f) ? rsqrtf(d) : 0.0f;  // in place: deg -> dinv
    }
}

// ------------------------------------------------- WMMA GEMM: h = x @ W ----
// One wave per 16x16 output tile. Block = 256 threads = 8 waves; wave w owns
// column tile n0 = 16*w (NHID=128 -> exactly 8 tiles). blockIdx.x owns 16 rows.
// V_WMMA_F32_16X16X4_F32 layouts (ISA 7.12.2):
//   A 16x4 : lane L holds M=L%16, K = kbase + (L/16)*2 + {0,1}  (2 VGPRs)
//   B 4x16 : lane L holds N=L%16, K = kbase + (L/16)*2 + {0,1}  (2 VGPRs)
//   C 16x16: vgpr v, lane L -> row v + (L/16)*8, col L%16       (8 VGPRs)
__global__ __launch_bounds__(256) void gcn_gemm_wmma(const float* __restrict__ x,
                                                     const float* __restrict__ W,
                                                     float* __restrict__ h, int N) {
    const int lane = threadIdx.x & 31;
    const int wave = threadIdx.x >> 5;
    const int m0   = blockIdx.x * 16;
    const int n0   = wave * 16;

    const int ml   = lane & 15;            // M (for A) / N (for B) within tile
    const int koff = (lane >> 4) * 2;      // K sub-offset for this half-wave

    int mi = m0 + ml;
    if (mi >= N) mi = N - 1;               // clamp (tail rows recomputed, harmless)

    const float* arow = x + (long long)mi * NFEAT;

    v8f c = {};
#pragma unroll
    for (int k0 = 0; k0 < NFEAT; k0 += 4) {
        const int k = k0 + koff;
        // A: two consecutive K values for this lane's row (8B-aligned)
        v2f a;
        a.x = arow[k];
        a.y = arow[k + 1];
        // B: two consecutive K rows, this lane's column
        v2f bb;
        bb.x = W[(long long)k * NHID + n0 + ml];
        bb.y = W[(long long)(k + 1) * NHID + n0 + ml];
        c = __builtin_amdgcn_wmma_f32_16x16x4_f32(
                /*neg_a=*/false, a, /*neg_b=*/false, bb,
                /*c_mod=*/(short)0, c, /*reuse_a=*/false, /*reuse_b=*/false);
    }

    const int colh  = n0 + ml;
    const int rbase = m0 + (lane >> 4) * 8;
#pragma unroll
    for (int v = 0; v < 8; ++v) {
        int r = rbase + v;
        if (r < N) h[(long long)r * NHID + colh] = c[v];
    }
}

// --------------------------------------- edge scatter: out[col] += norm*h[row]
// One wave per edge; lane owns 4 contiguous features (float4 gather,
// 4x global_atomic_add_f32 scatter).
__global__ __launch_bounds__(256) void gcn_scatter(const long long* __restrict__ row,
                                                   const long long* __restrict__ col,
                                                   const float* __restrict__ ew,
                                                   const float* __restrict__ dinv,
                                                   const float* __restrict__ h,
                                                   float* __restrict__ out, int E) {
    const int wid  = (int)(((long long)blockIdx.x * blockDim.x + threadIdx.x) >> 5);
    const int lane = threadIdx.x & 31;
    if (wid >= E) return;

    const long long r = row[wid];
    const long long c = col[wid];
    const float norm = dinv[(int)r] * sigmoidf_fast(ew[wid]) * dinv[(int)c];

    const float4 hv = *(const float4*)(h + r * NHID + lane * 4);
    float* op = out + c * NHID + lane * 4;
    fatomic_add(op + 0, hv.x * norm);
    fatomic_add(op + 1, hv.y * norm);
    fatomic_add(op + 2, hv.z * norm);
    fatomic_add(op + 3, hv.w * norm);
}

// ------------------------------------------- self-loop: out += h / deg ----
__global__ __launch_bounds__(256) void gcn_selfloop(const float* __restrict__ h,
                                                    const float* __restrict__ dinv,
                                                    float* __restrict__ out,
                                                    long long total) {
    long long i = (long long)blockIdx.x * blockDim.x + threadIdx.x;
    if (i >= total) return;
    int node = (int)(i >> 7);  // /NHID
    float di = dinv[node];
    out[i] += h[i] * di * di;  // self-edge norm = dinv*1.0*dinv
}

// --------------------------------------------------------------- launch ----
extern "C" void kernel_launch(void* const* d_in, const int* in_sizes, int n_in,
                              void* d_out, int out_size, void* d_ws, size_t ws_size,
                              hipStream_t stream) {
    const float*     x  = (const float*)d_in[0];
    const long long* ei = (const long long*)d_in[1];   // [2, E] int64
    const float*     ew = (const float*)d_in[2];
    const float*     W  = (const float*)d_in[3];
    const float*     b  = (const float*)d_in[4];
    float* out = (float*)d_out;

    const int N = in_sizes[0] / NFEAT;
    const int E = in_sizes[1] / 2;
    const long long total = (long long)N * NHID;

    const long long* row = ei;
    const long long* col = ei + E;

    float* deg = (float*)d_ws;          // N floats; becomes dinv in place
    float* h   = deg + N;               // N*NHID floats

    // 1) init deg=1 (self loop) and out=bias
    gcn_init_deg<<<(N + 255) / 256, 256, 0, stream>>>(deg, N);
    gcn_init_out_bias<<<(int)((total + 255) / 256), 256, 0, stream>>>(out, b, total);

    // 2) degree accumulation
    gcn_degree<<<(E + 255) / 256, 256, 0, stream>>>(col, ew, deg, E);

    // 3) dinv = rsqrt(deg)
    gcn_rsqrt<<<(N + 255) / 256, 256, 0, stream>>>(deg, N);

    // 4) h = x @ W via V_WMMA_F32_16X16X4_F32
    gcn_gemm_wmma<<<(N + 15) / 16, 256, 0, stream>>>(x, W, h, N);

    // 5) edge scatter (one wave per edge)
    {
        long long threads = (long long)E * 32;
        gcn_scatter<<<(int)((threads + 255) / 256), 256, 0, stream>>>(
            row, col, ew, deg, h, out, E);
    }

    // 6) self-loop contribution
    gcn_selfloop<<<(int)((total + 255) / 256), 256, 0, stream>>>(h, deg, out, total);
}